// model_6408091205990
// MI455X (gfx1250) — compile-verified
//
#include <hip/hip_runtime.h>

// GRU (I=1, H=5, O=3), T=10 steps, N = 4096*301 independent sequences.
// Strategy: 16 samples per wave32. Gate pre-activations computed as
//   D(16x16) = A(16x8 weights) x B(8x16 per-sample features [h0..h4, x, 1, 0])
// via two chained V_WMMA_F32_16X16X4_F32. Rows of D = gates (0-4:r, 5-9:z,
// 10-14: h_n = w_hh.h + b_hh), cols = samples. Elementwise GRU update done on
// VALU with per-lane full hidden copy; half-wave row split resolved with one
// ds_swizzle SWAPX16 per D register.

typedef __attribute__((ext_vector_type(2))) float v2f;
typedef __attribute__((ext_vector_type(8))) float v8f;

#define GRU_B 4096
#define GRU_T 10
#define GRU_S 301
#define GRU_N (GRU_B * GRU_S)   // 1,232,896 ; divisible by 16

__device__ __forceinline__ float swap_half(float v) {
  // lane L <-> lane L^16 (group-of-32 swizzle: and=0x1F, or=0, xor=0x10)
  int i = __builtin_bit_cast(int, v);
  i = __builtin_amdgcn_ds_swizzle(i, 0x401F);
  return __builtin_bit_cast(float, i);
}

__device__ __forceinline__ float tanh_fast(float x) {
#if __has_builtin(__builtin_amdgcn_tanhf)
  return __builtin_amdgcn_tanhf(x);            // v_tanh_f32 (gfx1250)
#else
  float e = __builtin_amdgcn_exp2f(2.88539008f * x);   // exp(2x)
  return 1.0f - 2.0f * __builtin_amdgcn_rcpf(e + 1.0f);
#endif
}

__device__ __forceinline__ float sigmoid_fast(float x) {
#if __has_builtin(__builtin_amdgcn_tanhf)
  return __builtin_fmaf(0.5f, __builtin_amdgcn_tanhf(0.5f * x), 0.5f);
#else
  float e = __builtin_amdgcn_exp2f(-1.44269504f * x);  // exp(-x)
  return __builtin_amdgcn_rcpf(1.0f + e);
#endif
}

__global__ __launch_bounds__(256) void gru_wmma_kernel(
    const float* __restrict__ x,     // (B, T, S, 1)
    const float* __restrict__ w_ih,  // (15, 1)
    const float* __restrict__ w_hh,  // (15, 5)
    const float* __restrict__ b_ih,  // (15,)
    const float* __restrict__ b_hh,  // (15,)
    const float* __restrict__ fc_w,  // (3, 5)
    const float* __restrict__ fc_b,  // (3,)
    float* __restrict__ out)         // (B, 3, S, 1)
{
  const int lane = threadIdx.x & 31;
  const int m    = lane & 15;        // sample-slot (cols) / gate-row (A) index
  const bool hi  = lane >= 16;

  const int wave = blockIdx.x * (blockDim.x >> 5) + (threadIdx.x >> 5);
  if (wave * 16 >= GRU_N) return;    // uniform per-wave guard (never partial)

  const int n = wave * 16 + m;       // sample id (lanes m and m+16 share it)
  const int b = n / GRU_S;
  const int s = n - b * GRU_S;
  const int xbase = b * (GRU_T * GRU_S) + s;   // x[b][t][s] = xbase + t*S

  // ---------------- A-matrix fragments (constant weights) ----------------
  // A layout (16x4 f32): lanes 0-15 rows M=0-15 with {K0,K1}; lanes 16-31 {K2,K3}.
  // A1: K=0..3 -> w_hh cols 0..3.  A2: K=4..7 -> [w_hh col4, w_ih, bias, 0].
  const int mm = (m < 15) ? m : 14;  // clamp unused row 15 (avoid OOB loads)
  float a10 = w_hh[mm * 5 + (hi ? 2 : 0)];
  float a11 = w_hh[mm * 5 + (hi ? 3 : 1)];
  float biasm = (mm < 10) ? (b_ih[mm] + b_hh[mm]) : b_hh[mm];
  float a20 = hi ? biasm : w_hh[mm * 5 + 4];
  float a21 = hi ? 0.0f : ((mm < 10) ? w_ih[mm] : 0.0f);
  if (m == 15) { a10 = 0.0f; a11 = 0.0f; a20 = 0.0f; a21 = 0.0f; }
  v2f A1; A1.x = a10; A1.y = a11;
  v2f A2; A2.x = a20; A2.y = a21;

  // Uniform scalars for the n-gate input path and the FC head (s_loads).
  float wn[5], bn[5];
#pragma unroll
  for (int j = 0; j < 5; ++j) { wn[j] = w_ih[10 + j]; bn[j] = b_ih[10 + j]; }

  // Per-sequence inputs (x is scalar per step).
  float xv[GRU_T];
#pragma unroll
  for (int t = 0; t < GRU_T; ++t) xv[t] = x[xbase + t * GRU_S];

  // Full hidden-state copy per lane.
  float h[5] = {0.f, 0.f, 0.f, 0.f, 0.f};

#pragma unroll
  for (int t = 0; t < GRU_T; ++t) {
    // B fragments (8x16 features): B layout mirrors A across half-waves:
    // lanes 0-15 hold {K0,K1} = (h0,h1); lanes 16-31 hold {K2,K3} = (h2,h3);
    // second fragment: lanes 0-15 {K4,K5} = (h4, x); lanes 16-31 {K6,K7} = (1, 0).
    v2f B1; B1.x = hi ? h[2] : h[0];
            B1.y = hi ? h[3] : h[1];
    v2f B2; B2.x = hi ? 1.0f : h[4];
            B2.y = hi ? 0.0f : xv[t];

    v8f c = {};
    c = __builtin_amdgcn_wmma_f32_16x16x4_f32(false, A1, false, B1,
                                              (short)0, c, false, false);
    c = __builtin_amdgcn_wmma_f32_16x16x4_f32(false, A2, false, B2,
                                              (short)0, c, false, false);

    // D layout: lane m holds rows 0-7, lane m+16 rows 8-15 (col = sample m).
    // One SWAPX16 per register gives each lane all 15 gate pre-activations.
    float p[15];
#pragma unroll
    for (int k = 0; k < 8; ++k) {
      float dk = c[k];
      float tk = swap_half(dk);
      p[k] = hi ? tk : dk;                   // rows 0..7
      if (k < 7) p[k + 8] = hi ? dk : tk;    // rows 8..14 (row 15 unused)
    }

    // GRU update (both half-waves compute the same sample — free in issue count).
#pragma unroll
    for (int j = 0; j < 5; ++j) {
      float r  = sigmoid_fast(p[j]);         // rows 0-4
      float z  = sigmoid_fast(p[5 + j]);     // rows 5-9
      float np = __builtin_fmaf(wn[j], xv[t], bn[j]);   // i_n
      np = __builtin_fmaf(r, p[10 + j], np);            // + r * h_n
      float nn = tanh_fast(np);
      h[j] = __builtin_fmaf(z, h[j] - nn, nn);          // (1-z)*n + z*h
    }
  }

  // FC head: y[o] = fc_b[o] + fc_w[o,:] . h ; only low half-wave stores.
  if (!hi) {
#pragma unroll
    for (int o = 0; o < 3; ++o) {
      float y = fc_b[o];
#pragma unroll
      for (int j = 0; j < 5; ++j) y = __builtin_fmaf(fc_w[o * 5 + j], h[j], y);
      out[(b * 3 + o) * GRU_S + s] = y;   // (B, 3, S, 1) layout
    }
  }
}

extern "C" void kernel_launch(void* const* d_in, const int* in_sizes, int n_in,
                              void* d_out, int out_size, void* d_ws, size_t ws_size,
                              hipStream_t stream) {
  (void)in_sizes; (void)n_in; (void)out_size; (void)d_ws; (void)ws_size;
  const float* x    = (const float*)d_in[0];
  const float* w_ih = (const float*)d_in[1];
  const float* w_hh = (const float*)d_in[2];
  const float* b_ih = (const float*)d_in[3];
  const float* b_hh = (const float*)d_in[4];
  const float* fc_w = (const float*)d_in[5];
  const float* fc_b = (const float*)d_in[6];
  float* out = (float*)d_out;

  const int threads = 256;                       // 8 waves/block
  const int samplesPerBlock = (threads / 32) * 16;   // 128
  const int blocks = (GRU_N + samplesPerBlock - 1) / samplesPerBlock;  // 9632
  gru_wmma_kernel<<<blocks, threads, 0, stream>>>(x, w_ih, w_hh, b_ih, b_hh,
                                                  fc_w, fc_b, out);
}